// VisibleRebuildModule_22591527977307
// MI455X (gfx1250) — compile-verified
//
#include <hip/hip_runtime.h>
#include <math.h>

#define S_DIM 256
#define NPIX (S_DIM * S_DIM)
#define NBATCH 64

typedef __attribute__((ext_vector_type(2))) float v2f;
typedef __attribute__((ext_vector_type(8))) float v8f;

// Sum across 16 lanes; both 16-lane halves hold identical data, so a
// butterfly with offsets 8..1 leaves the full sum in every lane.
__device__ __forceinline__ float butterfly16(float v) {
    v += __shfl_xor(v, 8, 32);
    v += __shfl_xor(v, 4, 32);
    v += __shfl_xor(v, 2, 32);
    v += __shfl_xor(v, 1, 32);
    return v;
}

// ---------------------------------------------------------------------------
// Kernel 1: one wave per batch. Solve for R_p (3x3) and T_p (3) and store
// 12 floats per batch into the workspace.
// ---------------------------------------------------------------------------
__global__ void __launch_bounds__(32)
pose_solve_kernel(const float* __restrict__ Offset,
                  const float* __restrict__ Posmap,
                  const float* __restrict__ mean,
                  const int* __restrict__ uv,
                  float* __restrict__ RT) {
    const int b    = blockIdx.x;
    const int lane = threadIdx.x & 31;
    const int q    = lane & 15;          // quad id, duplicated in both halves

    const float* offb = Offset + (size_t)b * 3 * NPIX;
    const float* posb = Posmap + (size_t)b * 3 * NPIX;

    float p[4][3], qd[4][3], col[4];
#pragma unroll
    for (int j = 0; j < 4; ++j) {
        const int k = 4 * q + j;         // keypoint 0..63 (within 68)
        const int r = uv[2 * k + 0];
        const int c = uv[2 * k + 1];
        col[j] = (float)c;
        const int pix = r * S_DIM + c;
#pragma unroll
        for (int d = 0; d < 3; ++d) {
            p[j][d]  = fmaf(offb[d * NPIX + pix], 6.0f, mean[d * NPIX + pix]);
            qd[j][d] = posb[d * NPIX + pix];
        }
    }

    // A[i][d] = p0[d] - p[i+1][d];  Bm[i][d] = q0[d] - q[i+1][d]
    float A[3][3], Bm[3][3];
#pragma unroll
    for (int i = 0; i < 3; ++i)
#pragma unroll
        for (int d = 0; d < 3; ++d) {
            A[i][d]  = p[0][d] - p[i + 1][d];
            Bm[i][d] = qd[0][d] - qd[i + 1][d];
        }

    // inv(A) via adjugate
    const float c00 = A[1][1] * A[2][2] - A[1][2] * A[2][1];
    const float c01 = A[1][2] * A[2][0] - A[1][0] * A[2][2];
    const float c02 = A[1][0] * A[2][1] - A[1][1] * A[2][0];
    const float det = A[0][0] * c00 + A[0][1] * c01 + A[0][2] * c02;
    const float id  = 1.0f / det;
    float inv[3][3];
    inv[0][0] = c00 * id;
    inv[0][1] = (A[0][2] * A[2][1] - A[0][1] * A[2][2]) * id;
    inv[0][2] = (A[0][1] * A[1][2] - A[0][2] * A[1][1]) * id;
    inv[1][0] = c01 * id;
    inv[1][1] = (A[0][0] * A[2][2] - A[0][2] * A[2][0]) * id;
    inv[1][2] = (A[0][2] * A[1][0] - A[0][0] * A[1][2]) * id;
    inv[2][0] = c02 * id;
    inv[2][1] = (A[0][1] * A[2][0] - A[0][0] * A[2][1]) * id;
    inv[2][2] = (A[0][0] * A[1][1] - A[0][1] * A[1][0]) * id;

    float R[3][3];
#pragma unroll
    for (int i = 0; i < 3; ++i)
#pragma unroll
        for (int d = 0; d < 3; ++d)
            R[i][d] = inv[i][0] * Bm[0][d] + inv[i][1] * Bm[1][d] + inv[i][2] * Bm[2][d];

    float T[3];
#pragma unroll
    for (int d = 0; d < 3; ++d) {
        float acc = 0.0f;
#pragma unroll
        for (int j = 0; j < 4; ++j)
            acc += qd[j][d] - (p[j][0] * R[0][d] + p[j][1] * R[1][d] + p[j][2] * R[2][d]);
        T[d] = acc * 0.25f;
    }

    // yaw only needs column 0 of the coarse (mean-over-quads) R
    const float rc00 = butterfly16(R[0][0]) * (1.0f / 16.0f);
    const float rc10 = butterfly16(R[1][0]) * (1.0f / 16.0f);
    const float rc20 = butterfly16(R[2][0]) * (1.0f / 16.0f);
    const float yaw  = atan2f(-rc20, sqrtf(rc00 * rc00 + rc10 * rc10));
    const float yr   = yaw * (1.0f / 3.14159265358979323846f);
    const float left  = fmaxf(yr * (float)S_DIM, 0.0f);
    const float right = fminf((float)S_DIM + yr * (float)S_DIM, (float)S_DIM);

    float w = 1.0f;
#pragma unroll
    for (int j = 0; j < 4; ++j)
        w *= ((col[j] <= left) || (col[j] >= right)) ? 0.001f : 1.0f;

    const float s = butterfly16(w);
    const float is = 1.0f / s;

    float Rp[3][3], Tp[3];
#pragma unroll
    for (int i = 0; i < 3; ++i)
#pragma unroll
        for (int d = 0; d < 3; ++d)
            Rp[i][d] = butterfly16(w * R[i][d]) * is;
#pragma unroll
    for (int d = 0; d < 3; ++d)
        Tp[d] = butterfly16(w * T[d]) * is;

    if (lane == 0) {
        float* o = RT + b * 12;
        o[0] = Rp[0][0]; o[1] = Rp[0][1]; o[2] = Rp[0][2];
        o[3] = Rp[1][0]; o[4] = Rp[1][1]; o[5] = Rp[1][2];
        o[6] = Rp[2][0]; o[7] = Rp[2][1]; o[8] = Rp[2][2];
        o[9] = Tp[0];    o[10] = Tp[1];   o[11] = Tp[2];
    }
}

// ---------------------------------------------------------------------------
// Kernel 2: streaming transform, 32 pixels per wave per iteration via two
// V_WMMA_F32_16X16X4_F32 ops.
//   A (16x4): rows 0..2 = [R_p^T | T_p], rest zero (per-batch constant)
//   B (4x16): columns = [x, y, z, 1]^T for 16 contiguous pixels
//   D rows 0..2 (lanes 0..15, vgprs 0..2) = output components, contiguous in N
// ---------------------------------------------------------------------------
__global__ void __launch_bounds__(256)
transform_kernel(const float* __restrict__ Offset,
                 const float* __restrict__ mean,
                 const float* __restrict__ RT,
                 float* __restrict__ out) {
    const int b    = blockIdx.y;
    const int lane = threadIdx.x & 31;
    const int wave = threadIdx.x >> 5;
    const int p0   = blockIdx.x * 256 + wave * 32;   // 32 pixels per wave
    const int ln   = lane & 15;
    const bool lo  = lane < 16;

    // Per-batch constant A matrix (branch-free; wmma needs EXEC all-ones)
    const float* rt = RT + b * 12;
    float a0 = 0.0f, a1 = 0.0f;
    if (ln < 3) {
        a0 = lo ? rt[ln]     : rt[6 + ln];   // R[0][c] | R[2][c]
        a1 = lo ? rt[3 + ln] : rt[9 + ln];   // R[1][c] | T[c]
    }
    v2f a; a.x = a0; a.y = a1;

    const float* offb = Offset + (size_t)b * 3 * NPIX;
    const int p = p0 + lane;

    // Streaming (non-temporal) loads of the big input; mean stays cached.
    const float x = __builtin_nontemporal_load(&offb[p]);
    const float y = __builtin_nontemporal_load(&offb[NPIX + p]);
    const float z = __builtin_nontemporal_load(&offb[2 * NPIX + p]);
    const float ax = fmaf(x, 6.0f, mean[p]);
    const float ay = fmaf(y, 6.0f, mean[NPIX + p]);
    const float az = fmaf(z, 6.0f, mean[2 * NPIX + p]);

    // Swap halves so each WMMA's B matrix sees its 16 pixels' x/y/z.
    const float sax = __shfl_xor(ax, 16, 32);
    const float say = __shfl_xor(ay, 16, 32);
    const float saz = __shfl_xor(az, 16, 32);

    // Group0 = pixels p0..p0+15, Group1 = pixels p0+16..p0+31
    v2f B0; B0.x = lo ? ax  : saz; B0.y = lo ? ay  : 1.0f;
    v2f B1; B1.x = lo ? sax : az;  B1.y = lo ? say : 1.0f;

    v8f cz = {};
    v8f d0 = __builtin_amdgcn_wmma_f32_16x16x4_f32(false, a, false, B0,
                                                   (short)0, cz, false, false);
    v8f d1 = __builtin_amdgcn_wmma_f32_16x16x4_f32(false, a, false, B1,
                                                   (short)0, cz, false, false);

    float* outb = out + (size_t)b * 3 * NPIX;
    const float d0c[3] = {d0[0], d0[1], d0[2]};
    const float d1c[3] = {d1[0], d1[1], d1[2]};
#pragma unroll
    for (int c = 0; c < 3; ++c) {
        const float g1 = __shfl_xor(d1c[c], 16, 32);   // group1 -> lanes 16..31
        const float v  = lo ? d0c[c] : g1;
        __builtin_nontemporal_store(v, &outb[c * NPIX + p]);
    }
}

extern "C" void kernel_launch(void* const* d_in, const int* in_sizes, int n_in,
                              void* d_out, int out_size, void* d_ws, size_t ws_size,
                              hipStream_t stream) {
    const float* Offset = (const float*)d_in[0];   // (64,3,256,256)
    const float* Posmap = (const float*)d_in[1];   // (64,3,256,256)
    const float* mean   = (const float*)d_in[2];   // (3,256,256)
    const int*   uv     = (const int*)d_in[3];     // (68,2)
    float* out = (float*)d_out;                    // (64,3,256,256)
    float* RT  = (float*)d_ws;                     // 64 * 12 floats

    pose_solve_kernel<<<NBATCH, 32, 0, stream>>>(Offset, Posmap, mean, uv, RT);

    dim3 grid(NPIX / 256, NBATCH);
    transform_kernel<<<grid, 256, 0, stream>>>(Offset, mean, RT, out);
}